// VQVAE_27797028339989
// MI455X (gfx1250) — compile-verified
//
#include <hip/hip_runtime.h>
#include <hip/hip_bf16.h>

typedef __attribute__((ext_vector_type(2))) float        v2f;
typedef __attribute__((ext_vector_type(8))) float        v8f;
typedef __attribute__((ext_vector_type(4))) unsigned int v4u;
typedef __attribute__((ext_vector_type(8))) int          v8i;

// LDS byte offset (workgroup-relative) of a __shared__ pointer.
#define LDS_OFF(p) ((unsigned)(unsigned long long)(__attribute__((address_space(3))) void*)(p))

// ---------------------------------------------------------------------------
// Tensor Data Mover: 1-D contiguous copy of `ndw` dwords global -> LDS.
// D# per CDNA5 ISA:
//   g0: [1:0]=count=1, [63:32]=lds_addr, [120:64]=global_addr, [127:126]=type=2
//   g1: [17:16]=data_size=2 (4B), tensor_dim0=[79:48], tensor_dim1=[111:80]=1,
//       tile_dim0=[127:112], tensor_dim0_stride=[207:160]
// Issued once per workgroup (wave 0); completion via s_wait_tensorcnt.
// ---------------------------------------------------------------------------
__device__ __forceinline__ void tdm_load_1d(const void* gsrc, unsigned lds_off,
                                            unsigned ndw) {
    unsigned long long ga = (unsigned long long)gsrc;
    v4u g0 = { 1u,                                   // count=1, is_restore=0
               lds_off,                              // lds_addr (bytes)
               (unsigned)ga,                         // global_addr[31:0]
               ((unsigned)(ga >> 32) & 0x01FFFFFFu)  // global_addr[56:32]
                   | 0x80000000u };                  // type=2 ("image")
    v8i g1 = { (int)0x00020000u,                     // data_size=2 (4 bytes)
               (int)((ndw & 0xFFFFu) << 16),         // tensor_dim0[15:0]
               (int)0x00010000u,                     // dim0 hi=0, tensor_dim1=1
               (int)((ndw & 0xFFFFu) << 16),         // dim1 hi=0, tile_dim0
               0,                                    // tile_dim1=0, tile_dim2=0
               (int)ndw,                             // tensor_dim0_stride[31:0]
               0, 0 };
    asm volatile("tensor_load_to_lds %0, %1" :: "s"(g0), "s"(g1) : "memory");
}

// ---------------------------------------------------------------------------
// conv1: [32,3,256,256] -> [32,16,128,128], k4 s2 p1, +bias, ReLU. Direct.
// ---------------------------------------------------------------------------
__global__ __launch_bounds__(256)
void vqvae_conv1(const float* __restrict__ imgs, const float* __restrict__ w1,
                 const float* __restrict__ b1, float* __restrict__ a1) {
    int idx = blockIdx.x * 256 + threadIdx.x;        // 32*16*128*128
    int wo = idx & 127;
    int ho = (idx >> 7) & 127;
    int co = (idx >> 14) & 15;
    int b  = idx >> 18;
    // warm the next input row segment while we compute this one
    __builtin_prefetch(&imgs[((b * 3) * 256 + (2 * ho + 3 < 256 ? 2 * ho + 3 : 255)) * 256 + 2 * wo], 0, 1);
    float acc = b1[co];
    for (int ci = 0; ci < 3; ++ci) {
#pragma unroll
        for (int kh = 0; kh < 4; ++kh) {
            int hi = 2 * ho - 1 + kh;
            if (hi < 0 || hi >= 256) continue;
#pragma unroll
            for (int kw = 0; kw < 4; ++kw) {
                int wi = 2 * wo - 1 + kw;
                if (wi < 0 || wi >= 256) continue;
                acc += imgs[((b * 3 + ci) * 256 + hi) * 256 + wi] *
                       w1[((co * 3 + ci) * 4 + kh) * 4 + kw];
            }
        }
    }
    a1[idx] = acc > 0.f ? acc : 0.f;
}

// ---------------------------------------------------------------------------
// conv2 as implicit GEMM on v_wmma_f32_16x16x4_f32.
// M = Cout = 32, K = Cin*kh*kw = 256, N = 32*64*64 pixels.
// Block: 128 threads (4 waves) -> 32 pixels x 32 Cout.
// Weights (32KB) arrive via Tensor Data Mover while VMEM stages the patch.
// ---------------------------------------------------------------------------
__global__ __launch_bounds__(128)
void vqvae_conv2_wmma(const float* __restrict__ a1, const float* __restrict__ w2,
                      const float* __restrict__ b2, float* __restrict__ z_e) {
    __shared__ float sw[32 * 256];    // w2: co*256 + (ci*16 + kh*4 + kw)
    __shared__ float sin[16 * 4 * 66];// input patch [ci][kh][x]

    const int tid = threadIdx.x;
    const int pix_base = blockIdx.x * 32;     // 4096 blocks * 32 pixels
    const int wo0 = pix_base & 63;            // 0 or 32
    const int ho  = (pix_base >> 6) & 63;
    const int b   = pix_base >> 12;

    if (tid < 32) tdm_load_1d(w2, LDS_OFF(sw), 32 * 256);   // async DMA, wave 0

    for (int i = tid; i < 16 * 4 * 66; i += 128) {
        int ci = i / (4 * 66);
        int r  = i % (4 * 66);
        int kh = r / 66;
        int x  = r % 66;
        int hi = 2 * ho - 1 + kh;
        int wi = 2 * wo0 - 1 + x;
        float v = 0.f;
        if (hi >= 0 && hi < 128 && wi >= 0 && wi < 128)
            v = a1[((b * 16 + ci) * 128 + hi) * 128 + wi];
        sin[i] = v;
    }
    if (tid < 32) __builtin_amdgcn_s_wait_tensorcnt(0);
    __syncthreads();

    const int wave  = tid >> 5;
    const int lane  = tid & 31;
    const int ln    = lane & 15;
    const int hk    = lane >> 4;      // K sub-offset select (ISA A/B layout)
    const int mhalf = wave & 1;
    const int ntile = wave >> 1;
    const int co_a  = mhalf * 16 + ln;

    v8f c = {};
    for (int k0 = 0; k0 < 256; k0 += 4) {
        v2f a, bf;
#pragma unroll
        for (int j = 0; j < 2; ++j) {
            int k  = k0 + 2 * hk + j;
            int ci = k >> 4;
            int kh = (k >> 2) & 3;
            int kw = k & 3;
            a[j]  = sw[co_a * 256 + k];
            bf[j] = sin[(ci * 4 + kh) * 66 + ntile * 32 + 2 * ln + kw];
        }
        c = __builtin_amdgcn_wmma_f32_16x16x4_f32(false, a, false, bf,
                                                  (short)0, c, false, false);
    }

    const int wo_base = wo0 + ntile * 16;
#pragma unroll
    for (int r = 0; r < 8; ++r) {
        int co = mhalf * 16 + r + 8 * hk;     // C/D rows: M=r (lanes 0-15), r+8 (16-31)
        float v = c[r] + b2[co];
        v = v > 0.f ? v : 0.f;
        z_e[((b * 32 + co) * 64 + ho) * 64 + wo_base + ln] = v;
    }
}

// ---------------------------------------------------------------------------
// codebook norms: cn[k] = sum_d cb[k][d]^2, K=512, D=32
// ---------------------------------------------------------------------------
__global__ __launch_bounds__(256)
void vqvae_cbnorm(const float* __restrict__ cb, float* __restrict__ cn) {
    int k = blockIdx.x * 256 + threadIdx.x;
    if (k < 512) {
        float s = 0.f;
        for (int d = 0; d < 32; ++d) { float v = cb[k * 32 + d]; s += v * v; }
        cn[k] = s;
    }
}

// ---------------------------------------------------------------------------
// VQ: argmin_k | ||z||^2 - ||c_k||^2 |  (degenerate metric of the reference),
// then encoded[b,:,h,w] = codebook[idx]. Norm table in LDS; fully coalesced.
// ---------------------------------------------------------------------------
__global__ __launch_bounds__(256)
void vqvae_vq(const float* __restrict__ z_e, const float* __restrict__ cn,
              const float* __restrict__ cb, float* __restrict__ enc) {
    __shared__ float sn[512];
    int tid = threadIdx.x;
    sn[tid] = cn[tid];
    sn[tid + 256] = cn[tid + 256];
    __syncthreads();

    int pix = blockIdx.x * 256 + tid;     // 131072 pixels
    int b  = pix >> 12;
    int hw = pix & 4095;

    float a = 0.f;
    for (int d = 0; d < 32; ++d) {
        float v = z_e[(b * 32 + d) * 4096 + hw];
        a += v * v;
    }
    float best = fabsf(a - sn[0]);
    int bi = 0;
    for (int k = 1; k < 512; ++k) {
        float dd = fabsf(a - sn[k]);
        if (dd < best) { best = dd; bi = k; }   // strict <: first-min == jnp.argmin
    }
    for (int d = 0; d < 32; ++d)
        enc[(b * 32 + d) * 4096 + hw] = cb[bi * 32 + d];
}

// ---------------------------------------------------------------------------
// deconv1 (ConvTranspose2d 32->16, k4 s2 p1) as WMMA GEMM per parity class.
// out[o] = sum in[i]*w[k], o = 2*i + k - 1 -> per (yo%2, xo%2) exactly 2x2 taps.
// K = Cin*2*2 = 128, M = 16. Block = 256 thr (8 waves) covers one output row.
// Weights via TDM, input row via VMEM staging.
// ---------------------------------------------------------------------------
__global__ __launch_bounds__(256)
void vqvae_deconv1_wmma(const float* __restrict__ enc, const float* __restrict__ dw1,
                        const float* __restrict__ db1, float* __restrict__ a2) {
    __shared__ float sw[32 * 16 * 16];  // dw1 [ci][co][kh][kw] as-is
    __shared__ float sin[32 * 2 * 66];  // [ci][y-tap slot][xi+1], xi in -1..64

    const int tid = threadIdx.x;
    const int row = blockIdx.x;         // b*128 + yo, 4096 blocks
    const int yo  = row & 127;
    const int b   = row >> 7;
    const int py  = yo & 1;

    if (tid < 32) tdm_load_1d(dw1, LDS_OFF(sw), 32 * 256);   // async DMA, wave 0

    for (int i = tid; i < 32 * 2 * 66; i += 256) {
        int ci = i / 132;
        int r  = i % 132;
        int s  = r / 66;
        int x  = r % 66;
        int kh = (py ? 0 : 1) + 2 * s;        // valid kh taps for this yo parity
        int yi = (yo + 1 - kh) >> 1;
        int xi = x - 1;
        float v = 0.f;
        if (yi >= 0 && yi < 64 && xi >= 0 && xi < 64)
            v = enc[((b * 32 + ci) * 64 + yi) * 64 + xi];
        sin[i] = v;
    }
    if (tid < 32) __builtin_amdgcn_s_wait_tensorcnt(0);
    __syncthreads();

    const int wave = tid >> 5;          // 0..7
    const int lane = tid & 31;
    const int ln = lane & 15;
    const int hk = lane >> 4;
    const int px = wave & 1;            // x parity of this tile
    const int g  = wave >> 1;           // 16-pixel group 0..3
    // pixel column ln -> xo = px + 2*(g*16 + ln)

    v8f c = {};
    for (int k0 = 0; k0 < 128; k0 += 4) {
        v2f a, bf;
#pragma unroll
        for (int j = 0; j < 2; ++j) {
            int k  = k0 + 2 * hk + j;
            int ci = k >> 2;
            int s  = (k >> 1) & 1;
            int u  = k & 1;
            int kh = (py ? 0 : 1) + 2 * s;
            int kw = (px ? 0 : 1) + 2 * u;
            a[j] = sw[((ci * 16 + ln) * 4 + kh) * 4 + kw];
            // xi per column: px=0: u=0 -> m, u=1 -> m-1 ; px=1: u=0 -> m+1, u=1 -> m
            int xi = g * 16 + ln + (px ? (u ? 0 : 1) : (u ? -1 : 0));
            bf[j] = sin[(ci * 2 + s) * 66 + xi + 1];
        }
        c = __builtin_amdgcn_wmma_f32_16x16x4_f32(false, a, false, bf,
                                                  (short)0, c, false, false);
    }

#pragma unroll
    for (int r = 0; r < 8; ++r) {
        int co = r + 8 * hk;
        float v = c[r] + db1[co];
        v = v > 0.f ? v : 0.f;
        int xo = px + 2 * (g * 16 + ln);
        a2[((b * 16 + co) * 128 + yo) * 128 + xo] = v;
    }
}

// ---------------------------------------------------------------------------
// deconv2 (16->3, k4 s2 p1) + ReLU, direct (M=3 not WMMA friendly; 64 MACs/out).
// ---------------------------------------------------------------------------
__global__ __launch_bounds__(256)
void vqvae_deconv2(const float* __restrict__ a2, const float* __restrict__ dw2,
                   const float* __restrict__ db2, float* __restrict__ dec) {
    int idx = blockIdx.x * 256 + threadIdx.x;   // 32*3*256*256
    int xo = idx & 255;
    int yo = (idx >> 8) & 255;
    int co = (idx >> 16) % 3;
    int b  = idx / (3 * 256 * 256);
    float acc = db2[co];
#pragma unroll
    for (int jy = 0; jy < 2; ++jy) {
        int kh = ((yo & 1) ? 0 : 1) + 2 * jy;
        int yi = (yo + 1 - kh) >> 1;
        if (yi < 0 || yi >= 128) continue;
#pragma unroll
        for (int jx = 0; jx < 2; ++jx) {
            int kw = ((xo & 1) ? 0 : 1) + 2 * jx;
            int xi = (xo + 1 - kw) >> 1;
            if (xi < 0 || xi >= 128) continue;
            for (int ci = 0; ci < 16; ++ci)
                acc += a2[((b * 16 + ci) * 128 + yi) * 128 + xi] *
                       dw2[((ci * 3 + co) * 4 + kh) * 4 + kw];
        }
    }
    dec[idx] = acc > 0.f ? acc : 0.f;
}

// ---------------------------------------------------------------------------
extern "C" void kernel_launch(void* const* d_in, const int* in_sizes, int n_in,
                              void* d_out, int out_size, void* d_ws, size_t ws_size,
                              hipStream_t stream) {
    const float* imgs = (const float*)d_in[0];
    const float* w1   = (const float*)d_in[1];
    const float* b1   = (const float*)d_in[2];
    const float* w2   = (const float*)d_in[3];
    const float* b2   = (const float*)d_in[4];
    const float* cb   = (const float*)d_in[5];
    const float* dw1  = (const float*)d_in[6];
    const float* db1  = (const float*)d_in[7];
    const float* dw2  = (const float*)d_in[8];
    const float* db2  = (const float*)d_in[9];

    float* z_e = (float*)d_out;                 // [32,32,64,64]  = 4194304
    float* enc = z_e + 4194304;                 // [32,32,64,64]  = 4194304
    float* dec = z_e + 8388608;                 // [32,3,256,256] = 6291456

    float* a1 = (float*)d_ws;                   // [32,16,128,128] = 8388608 floats
    float* cn = a1 + 8388608;                   // 512 codebook norms
    // a1 is dead after conv2 -> deconv1 output reuses the same buffer.

    vqvae_conv1<<<32768, 256, 0, stream>>>(imgs, w1, b1, a1);
    vqvae_conv2_wmma<<<4096, 128, 0, stream>>>(a1, w2, b2, z_e);
    vqvae_cbnorm<<<2, 256, 0, stream>>>(cb, cn);
    vqvae_vq<<<512, 256, 0, stream>>>(z_e, cn, cb, enc);
    vqvae_deconv1_wmma<<<4096, 256, 0, stream>>>(enc, dw1, db1, a1);
    vqvae_deconv2<<<24576, 256, 0, stream>>>(a1, dw2, db2, dec);
}